// TensorSquare_75196287418996
// MI455X (gfx1250) — compile-verified
//
#include <hip/hip_runtime.h>
#include <math.h>

typedef float v2f __attribute__((ext_vector_type(2)));
typedef float v8f __attribute__((ext_vector_type(8)));

#define KSTEPS 84      // K = 336 = 84 * 4   (325 symmetric pairs, padded)
#define NTILES 21      // N = 336 = 21 * 16  (325 output cols, padded)
#define NCOLS  325
#define WAVES_PER_BLOCK 8

// ---------------- constexpr pair-index table (I,J per K element) -------------
// Pair order: p = i*25 - i*(i-1)/2 + (j-i) for 0 <= i <= j < 25.
// Entry v[2*kk + h] packs (I,J) for K rows (4kk+2h) and (4kk+2h+1), 5 bits each.
struct IJTab { unsigned v[2 * KSTEPS]; };
constexpr IJTab makeTab() {
  IJTab t{};
  unsigned I[4 * KSTEPS] = {}, J[4 * KSTEPS] = {};
  int p = 0;
  for (unsigned a = 0; a < 25; ++a)
    for (unsigned b = a; b < 25; ++b) { I[p] = a; J[p] = b; ++p; }
  for (; p < 4 * KSTEPS; ++p) { I[p] = 0; J[p] = 0; }
  for (int q = 0; q < 2 * KSTEPS; ++q) {
    int p0 = 2 * q;
    t.v[q] = I[p0] | (J[p0] << 5) | (I[p0 + 1] << 10) | (J[p0 + 1] << 15);
  }
  return t;
}
__constant__ IJTab IJTAB = makeTab();

// ---------------- builder: Wigner-3j coefficient matrix ----------------------
__device__ __constant__ double FACT[18] = {
  1., 1., 2., 6., 24., 120., 720., 5040., 40320., 362880., 3628800.,
  39916800., 479001600., 6227020800., 87178291200., 1307674368000.,
  20922789888000., 355687428096000.};

struct cplx { double re, im; };
__device__ inline cplx cmul(cplx a, cplx b) {
  return {a.re * b.re - a.im * b.im, a.re * b.im + a.im * b.re};
}
__device__ inline cplx cadd(cplx a, cplx b) { return {a.re + b.re, a.im + b.im}; }
__device__ inline cplx cconj(cplx a) { return {a.re, -a.im}; }
__device__ inline cplx cscale(cplx a, double s) { return {a.re * s, a.im * s}; }
__device__ inline int imax2(int a, int b) { return a > b ? a : b; }
__device__ inline int imin2(int a, int b) { return a < b ? a : b; }

__device__ double su2_cg(int j1, int j2, int j3, int m1, int m2, int m3) {
  if (m1 + m2 != m3) return 0.0;
  double pref = sqrt((2.0 * j3 + 1.0) * FACT[j1 + j2 - j3] * FACT[j1 - j2 + j3] *
                     FACT[-j1 + j2 + j3] / FACT[j1 + j2 + j3 + 1] *
                     FACT[j3 + m3] * FACT[j3 - m3] * FACT[j1 + m1] *
                     FACT[j1 - m1] * FACT[j2 + m2] * FACT[j2 - m2]);
  int kmin = imax2(0, imax2(j2 - j3 - m1, j1 - j3 + m2));
  int kmax = imin2(j1 + j2 - j3, imin2(j1 - m1, j2 + m2));
  double s = 0.0;
  for (int k = kmin; k <= kmax; ++k) {
    double d = FACT[k] * FACT[j1 + j2 - j3 - k] * FACT[j1 - m1 - k] *
               FACT[j2 + m2 - k] * FACT[j3 - j2 + m1 + k] * FACT[j3 - j1 - m2 + k];
    s += ((k & 1) ? -1.0 : 1.0) / d;
  }
  return pref * s;
}

// real->complex spherical harmonic change of basis (e3nn), row-major (2l+1)^2
__device__ void buildq(int l, cplx* q) {
  const int d = 2 * l + 1;
  for (int a = 0; a < d * d; ++a) q[a] = {0.0, 0.0};
  const double s = 0.7071067811865475244;
  for (int m = -l; m <= -1; ++m) {
    q[(l + m) * d + (l - m)] = {s, 0.0};
    q[(l + m) * d + (l + m)] = {0.0, -s};
  }
  q[l * d + l] = {1.0, 0.0};
  for (int m = 1; m <= l; ++m) {
    double sg = (m & 1) ? -1.0 : 1.0;
    q[(l + m) * d + (l + m)] = {sg * s, 0.0};
    q[(l + m) * d + (l - m)] = {0.0, sg * s};
  }
  cplx f;
  switch (l & 3) {  // (-i)^l
    case 0: f = {1.0, 0.0}; break;
    case 1: f = {0.0, -1.0}; break;
    case 2: f = {-1.0, 0.0}; break;
    default: f = {0.0, 1.0}; break;
  }
  for (int a = 0; a < d * d; ++a) q[a] = cmul(f, q[a]);
}

// write one value of M into the WMMA-B swizzled buffer.
// B 4x16 layout (wave32): VGPR0 = {K0 lanes0-15, K2 lanes16-31}, VGPR1 = {K1, K3}
__device__ inline void storeB(float* B, int p, int n, float v) {
  int kk = p >> 2, rem = p & 3;
  int h = rem >> 1, t = rem & 1;
  int nt = n >> 4, cc = n & 15;
  int lane = h * 16 + cc;
  B[(((size_t)kk * NTILES + nt) * 32 + lane) * 2 + t] = v;
}

__global__ __launch_bounds__(64)
void tsq_build(float* __restrict__ Bswz) {
  const int tid = threadIdx.x;
  const int total = KSTEPS * NTILES * 32 * 2;
  for (int i = tid; i < total; i += 64) Bswz[i] = 0.0f;
  __syncthreads();

  // ---- enumerate chunks in the reference's generation order ----
  int cl1[45], cl2[45], cl3[45], cs1[45], cs2[45], cuvw[45], cp3[45];
  const int Ls[5]  = {0, 1, 2, 3, 4};
  const int Ps[5]  = {1, -1, 1, -1, 1};
  const int Offs[5] = {0, 1, 4, 9, 16};
  int nc = 0;
  for (int i1 = 0; i1 < 5; ++i1) {
    for (int i2 = 0; i2 < 5; ++i2) {
      const int l1 = Ls[i1], l2 = Ls[i2];
      if (i1 < i2) {
        int lo = (l1 > l2) ? l1 - l2 : l2 - l1;
        for (int l3 = lo; l3 <= l1 + l2; ++l3) {
          cl1[nc] = l1; cl2[nc] = l2; cl3[nc] = l3;
          cs1[nc] = Offs[i1]; cs2[nc] = Offs[i2];
          cuvw[nc] = 1; cp3[nc] = Ps[i1] * Ps[i2]; ++nc;
        }
      } else if (i1 == i2) {
        for (int l3 = 0; l3 <= 2 * l1; l3 += 2) {
          cl1[nc] = l1; cl2[nc] = l2; cl3[nc] = l3;
          cs1[nc] = Offs[i1]; cs2[nc] = Offs[i1];
          cuvw[nc] = 0; cp3[nc] = Ps[i1] * Ps[i2]; ++nc;
        }
      }
    }
  }
  if (tid >= nc) return;
  const int c = tid;

  // ---- sorted output offset: key = (l, -p*(-1)^l, orig index) ----
  const int l3c = cl3[c];
  const int sc = -cp3[c] * ((l3c % 2 == 0) ? 1 : -1);
  int col0 = 0;
  for (int o = 0; o < nc; ++o) {
    if (o == c) continue;
    int so = -cp3[o] * ((cl3[o] % 2 == 0) ? 1 : -1);
    bool less = (cl3[o] < l3c) ||
                (cl3[o] == l3c && (so < sc || (so == sc && o < c)));
    if (less) col0 += 2 * cl3[o] + 1;
  }

  // ---- real Wigner-3j for this chunk ----
  const int l1 = cl1[c], l2 = cl2[c], l3 = cl3[c];
  const int d1 = 2 * l1 + 1, d2 = 2 * l2 + 1, d3 = 2 * l3 + 1;

  double C[9][9][17];
  for (int i = 0; i < d1; ++i)
    for (int k = 0; k < d2; ++k)
      for (int m = 0; m < d3; ++m) C[i][k][m] = 0.0;
  for (int m1 = -l1; m1 <= l1; ++m1)
    for (int m2 = -l2; m2 <= l2; ++m2) {
      int m3 = m1 + m2;
      if (m3 >= -l3 && m3 <= l3)
        C[l1 + m1][l2 + m2][l3 + m3] = su2_cg(l1, l2, l3, m1, m2, m3);
    }

  cplx q1[81], q2[81], q3[289];
  buildq(l1, q1); buildq(l2, q2); buildq(l3, q3);

  double CcR[9][9][17], CcI[9][9][17];
  double nR = 0.0, nI = 0.0;
  for (int a = 0; a < d1; ++a) {
    for (int b = 0; b < d2; ++b) {
      cplx t[17];
      for (int m = 0; m < d3; ++m) t[m] = {0.0, 0.0};
      for (int i = 0; i < d1; ++i) {
        cplx qa = q1[i * d1 + a];
        if (qa.re == 0.0 && qa.im == 0.0) continue;
        for (int k = 0; k < d2; ++k) {
          cplx qb = cmul(qa, q2[k * d2 + b]);
          if (qb.re == 0.0 && qb.im == 0.0) continue;
          for (int m = 0; m < d3; ++m) {
            double cv = C[i][k][m];
            if (cv != 0.0) t[m] = cadd(t[m], cscale(qb, cv));
          }
        }
      }
      for (int n = 0; n < d3; ++n) {
        cplx s = {0.0, 0.0};
        for (int m = 0; m < d3; ++m)
          s = cadd(s, cmul(cconj(q3[m * d3 + n]), t[m]));
        CcR[a][b][n] = s.re; CcI[a][b][n] = s.im;
        nR += s.re * s.re; nI += s.im * s.im;
      }
    }
  }
  const bool useR = (nR >= nI);
  const double norm = sqrt(useR ? nR : nI);
  const double alpha = cuvw[c] ? (double)d3
                     : (l3 == 0 ? (double)d3 / (double)(d1 + 2) : (double)d3 * 0.5);
  const double scale = sqrt(alpha) / norm;

  // ---- emit into symmetric-pair rows ----
  const int s1 = cs1[c], s2 = cs2[c];
  if (cuvw[c]) {
    for (int a = 0; a < d1; ++a)
      for (int b = 0; b < d2; ++b) {
        int gi = s1 + a, gj = s2 + b;             // gi < gj guaranteed
        int p = gi * 25 - gi * (gi - 1) / 2 + (gj - gi);
        for (int k = 0; k < d3; ++k) {
          double v = (useR ? CcR[a][b][k] : CcI[a][b][k]) * scale;
          storeB(Bswz, p, col0 + k, (float)v);
        }
      }
  } else {  // UUU: fold symmetric contributions x_a*x_b*(cg[a,b]+cg[b,a])
    for (int a = 0; a < d1; ++a)
      for (int b = a; b < d1; ++b) {
        int gi = s1 + a, gj = s1 + b;
        int p = gi * 25 - gi * (gi - 1) / 2 + (gj - gi);
        for (int k = 0; k < d3; ++k) {
          double v = (useR ? CcR[a][b][k] : CcI[a][b][k]);
          if (a != b) v += (useR ? CcR[b][a][k] : CcI[b][a][k]);
          storeB(Bswz, p, col0 + k, (float)(v * scale));
        }
      }
  }
}

// ---------------- main kernel: Y(16x336) @ M(336x336) via WMMA f32 ----------
__global__ __launch_bounds__(32 * WAVES_PER_BLOCK)
void tsq_main(const float* __restrict__ x, const float* __restrict__ Bswz,
              float* __restrict__ out) {
  __shared__ float sxall[WAVES_PER_BLOCK * 16 * 25];
  const int lane = threadIdx.x & 31;
  const int wave = threadIdx.x >> 5;
  const long tile = (long)blockIdx.x * WAVES_PER_BLOCK + wave;
  const long rowBase = tile * 16;

  // stage this wave's 16x25 input tile into LDS (coalesced)
  float* sx = &sxall[wave * 400];
  const float* xg = x + rowBase * 25;
  for (int idx = lane; idx < 400; idx += 32) sx[idx] = xg[idx];
  __syncthreads();

  const int h = lane >> 4;         // K-half select (A layout: K0,K1 | K2,K3)
  const int cidx = lane & 15;      // A row (M) == C col (N) index
  const float* srow = sx + cidx * 25;

  for (int g = 0; g < 3; ++g) {    // 21 N-tiles in 3 groups of 7
    v8f acc[7];
#pragma unroll
    for (int t = 0; t < 7; ++t) acc[t] = (v8f){0.f, 0.f, 0.f, 0.f, 0.f, 0.f, 0.f, 0.f};

    for (int kk = 0; kk < KSTEPS; ++kk) {
      // A fragment: pair products Y[row, 4kk+2h .. 4kk+2h+1]
      unsigned e = IJTAB.v[2 * kk + h];
      v2f A;
      A.x = srow[e & 31] * srow[(e >> 5) & 31];
      A.y = srow[(e >> 10) & 31] * srow[(e >> 15) & 31];

      const v2f* bp = (const v2f*)Bswz + ((size_t)kk * NTILES + g * 7) * 32 + lane;
#pragma unroll
      for (int t = 0; t < 7; ++t) {
        v2f B = bp[(size_t)t * 32];
        acc[t] = __builtin_amdgcn_wmma_f32_16x16x4_f32(
            false, A, false, B, (short)0, acc[t], false, false);
      }
    }

    // store C/D: VGPR r holds rows r (lanes 0-15) and r+8 (lanes 16-31)
#pragma unroll
    for (int t = 0; t < 7; ++t) {
      int n0 = (g * 7 + t) * 16 + cidx;
      if (n0 < NCOLS) {
#pragma unroll
        for (int r = 0; r < 8; ++r) {
          long row = rowBase + r + 8 * h;
          out[row * NCOLS + n0] = acc[t][r];
        }
      }
    }
  }
}

extern "C" void kernel_launch(void* const* d_in, const int* in_sizes, int n_in,
                              void* d_out, int out_size, void* d_ws, size_t ws_size,
                              hipStream_t stream) {
  (void)n_in; (void)out_size; (void)ws_size;
  const float* x = (const float*)d_in[0];
  float* out = (float*)d_out;
  float* Bswz = (float*)d_ws;   // 84*21*32*2 floats = 451,584 bytes

  tsq_build<<<1, 64, 0, stream>>>(Bswz);

  const int nrows = in_sizes[0] / 25;          // 262144
  const int tiles = nrows / 16;                // 16384
  const int blocks = tiles / WAVES_PER_BLOCK;  // 2048
  tsq_main<<<blocks, 32 * WAVES_PER_BLOCK, 0, stream>>>(x, Bswz, out);
}